// PostProcess_caltech_76871324664096
// MI455X (gfx1250) — compile-verified
//
#include <hip/hip_runtime.h>
#include <math.h>

// ---------------- problem constants ----------------
constexpr int NB   = 256;     // batches
constexpr int NQ   = 16384;   // queries
constexpr int NTOP = 300;     // top-k
constexpr int NPAD = 304;     // 19*16, padded for 16x16 WMMA tiles
constexpr int NTIL = 19;      // tiles per dim
constexpr int NW   = 10;      // 32-bit mask words per row (ceil(300/32))
constexpr float IOU_THR = 0.7f;

constexpr int TPB1 = 1024;        // top-k block (32 waves)
constexpr int EPT  = NQ / TPB1;   // 16 keys per thread
constexpr int TPB2 = 320;         // NMS block (10 waves)

// ---------------- CDNA5 vector types ----------------
typedef float    v2f  __attribute__((ext_vector_type(2)));
typedef float    v8f  __attribute__((ext_vector_type(8)));
typedef unsigned u32x4 __attribute__((ext_vector_type(4)));
typedef int      i32x4 __attribute__((ext_vector_type(4)));
typedef int      i32x8 __attribute__((ext_vector_type(8)));

// =====================================================================
// Kernel 1: per-batch top-300 via in-register radix select (32-bit keys,
// monotone map of l1-l0), bitonic sort of the 300 winners, box transform.
// =====================================================================
__global__ __launch_bounds__(TPB1) void topk_kernel(
    const float* __restrict__ logits,   // [B,Q,2]
    const float* __restrict__ pboxes,   // [B,Q,4] cxcywh
    const float* __restrict__ tsizes,   // [B,2]  (h,w)
    float* __restrict__ ws_scores,      // [B,NTOP]
    float* __restrict__ ws_boxes)       // [B,NTOP,4] xyxy scaled
{
  __shared__ int red[32];
  __shared__ int s_total;
  __shared__ int s_cnt_sel, s_cnt_tie;
  __shared__ unsigned long long sel[512];

  const int b    = blockIdx.x;
  const int t    = threadIdx.x;
  const int lane = t & 31;
  const int wid  = t >> 5;
  const float* lg = logits + (size_t)b * NQ * 2;

  // Load 16 score keys per thread; map float -> order-preserving uint.
  unsigned keys[EPT];
#pragma unroll
  for (int i = 0; i < EPT; ++i) {
    const int q = t * EPT + i;
    const float2 l = *(const float2*)(lg + (size_t)q * 2);
    const float s = l.y - l.x;                 // monotone in softmax prob of class 1
    const unsigned u = __float_as_uint(s);
    keys[i] = (u & 0x80000000u) ? ~u : (u | 0x80000000u);
  }
  // Warm the box gather path (global_prefetch_b8 on gfx1250).
  __builtin_prefetch(pboxes + (size_t)b * NQ * 4 + (size_t)t * 64, 0, 1);

  // Radix-select the 300th-largest key: 32 rounds of block-wide counting.
  unsigned prefix = 0;
  int r = NTOP;
  for (int bit = 31; bit >= 0; --bit) {
    const unsigned hi = (prefix | (1u << bit)) >> bit;
    int c = 0;
#pragma unroll
    for (int i = 0; i < EPT; ++i) c += (int)((keys[i] >> bit) == hi);
    for (int off = 16; off > 0; off >>= 1) c += __shfl_down(c, off, 32);
    if (lane == 0) red[wid] = c;
    __syncthreads();
    if (wid == 0) {
      int v = red[lane];
      for (int off = 16; off > 0; off >>= 1) v += __shfl_down(v, off, 32);
      if (lane == 0) s_total = v;
    }
    __syncthreads();
    const int total = s_total;       // uniform
    if (total >= r) prefix |= (1u << bit);
    else            r -= total;
    __syncthreads();                 // protect s_total for next round
  }
  // prefix == 300th-largest key value; r == #ties of that value still needed.

  if (t == 0) { s_cnt_sel = 0; s_cnt_tie = 0; }
  if (t < 512) sel[t] = 0ull;        // pad entries sort to the end
  __syncthreads();

#pragma unroll
  for (int i = 0; i < EPT; ++i) {
    const unsigned k = keys[i];
    int pos = -1;
    if (k > prefix) {
      pos = atomicAdd(&s_cnt_sel, 1);
    } else if (k == prefix) {
      const int tp = atomicAdd(&s_cnt_tie, 1);
      if (tp < r) pos = atomicAdd(&s_cnt_sel, 1);
    }
    if (pos >= 0 && pos < NTOP) {
      const unsigned idx = (unsigned)(t * EPT + i);
      sel[pos] = ((unsigned long long)k << 32) |
                 (unsigned long long)(0xFFFFFFFFu - idx);   // key desc, idx asc
    }
  }
  __syncthreads();

  // Bitonic sort 512 entries, descending.
  for (int k = 2; k <= 512; k <<= 1) {
    for (int j = k >> 1; j > 0; j >>= 1) {
      if (t < 512) {
        const int ixj = t ^ j;
        if (ixj > t) {
          const unsigned long long a  = sel[t];
          const unsigned long long c2 = sel[ixj];
          const bool up = ((t & k) == 0);
          if (up ? (a < c2) : (a > c2)) { sel[t] = c2; sel[ixj] = a; }
        }
      }
      __syncthreads();
    }
  }

  // Emit rank-ordered scores + transformed boxes for ranks 0..299.
  if (t < NTOP) {
    const unsigned long long e = sel[t];
    const unsigned k   = (unsigned)(e >> 32);
    const unsigned idx = 0xFFFFFFFFu - (unsigned)(e & 0xFFFFFFFFu);
    const unsigned u   = (k & 0x80000000u) ? (k ^ 0x80000000u) : ~k;
    const float sdiff  = __uint_as_float(u);
    const float score  = 1.0f / (1.0f + __expf(-sdiff));   // softmax prob class 1

    const float4 cb = *(const float4*)(pboxes + ((size_t)b * NQ + idx) * 4);
    const float img_h = tsizes[b * 2 + 0];
    const float img_w = tsizes[b * 2 + 1];
    float* ob = ws_boxes + ((size_t)b * NTOP + t) * 4;
    ob[0] = (cb.x - 0.5f * cb.z) * img_w;
    ob[1] = (cb.y - 0.5f * cb.w) * img_h;
    ob[2] = (cb.x + 0.5f * cb.z) * img_w;
    ob[3] = (cb.y + 0.5f * cb.w) * img_h;
    ws_scores[(size_t)b * NTOP + t] = score;
  }
}

// =====================================================================
// Kernel 2: per-batch NMS. TDM stages the 300x4 box tile into LDS;
// pairwise union base (area_i + area_j) per 16x16 tile via
// v_wmma_f32_16x16x4_f32; suppression bitmask + sequential keep loop.
// =====================================================================
__global__ __launch_bounds__(TPB2) void nms_kernel(
    const float* __restrict__ ws_boxes,   // [B,NTOP,4]
    const float* __restrict__ ws_scores,  // [B,NTOP]
    float* __restrict__ out)              // scores | labels | boxes | keep
{
  __shared__ __align__(16) float s_box[NPAD * 4];
  __shared__ float    s_area[NPAD];
  __shared__ unsigned s_mask[NPAD * NW];
  __shared__ unsigned s_keep[NW];

  const int b    = blockIdx.x;
  const int t    = threadIdx.x;
  const int lane = t & 31;
  const int wid  = t >> 5;

  // Zero suppression mask and the 4 padding boxes.
  for (int i = t; i < NPAD * NW; i += TPB2) s_mask[i] = 0u;
  if (t < (NPAD - NTOP) * 4) s_box[NTOP * 4 + t] = 0.0f;

  // ---- TDM: DMA 1200 contiguous f32 (300 boxes) into LDS, wave 0 only ----
  if (t < 32) {
    const unsigned long long ga =
        (unsigned long long)(const void*)(ws_boxes + (size_t)b * NTOP * 4);
    const unsigned lds = (unsigned)(unsigned long long)(const void*)&s_box[0];
    const unsigned n = NTOP * 4;     // 1200 elements, data_size = 4B

    u32x4 g0;
    g0[0] = 1u;                                        // count=1 (valid D#)
    g0[1] = lds;                                       // lds_addr (bytes)
    g0[2] = (unsigned)(ga & 0xFFFFFFFFull);            // global_addr[31:0]
    g0[3] = (unsigned)((ga >> 32) & 0x1FFFFFFull)      // global_addr[56:32]
            | (2u << 30);                              // type=2 ("image")
    i32x8 g1;
    g1[0] = (int)(2u << 16);                           // data_size=2 -> 4 bytes
    g1[1] = (int)((n & 0xFFFFu) << 16);                // tensor_dim0[15:0]
    g1[2] = (int)((n >> 16) | (1u << 16));             // tensor_dim0 hi | tensor_dim1=1
    g1[3] = (int)(n << 16);                            // tensor_dim1 hi=0 | tile_dim0=n
    g1[4] = 0;                                         // tile_dim1=0, tile_dim2=0
    g1[5] = (int)n;                                    // tensor_dim0_stride
    g1[6] = 0;
    g1[7] = 0;
    i32x4 gz  = {0, 0, 0, 0};
    i32x8 gz8 = {0, 0, 0, 0, 0, 0, 0, 0};
    __builtin_amdgcn_tensor_load_to_lds(g0, g1, gz, gz, gz8, 0);
    __builtin_amdgcn_s_wait_tensorcnt(0);
  }
  __syncthreads();

  // Areas (clamped like the reference).
  if (t < NPAD) {
    const float x0 = s_box[t * 4 + 0], y0 = s_box[t * 4 + 1];
    const float x1 = s_box[t * 4 + 2], y1 = s_box[t * 4 + 3];
    s_area[t] = fmaxf(x1 - x0, 0.0f) * fmaxf(y1 - y0, 0.0f);
  }
  __syncthreads();

  // ---- pairwise IoU tiles: union base via WMMA, rest VALU ----
  for (int tile = wid; tile < NTIL * NTIL; tile += NW) {
    const int ti = tile / NTIL, tj = tile % NTIL;
    const int m0 = ti * 16, n0 = tj * 16;
    const bool lo = lane < 16;

    // A (16x4): rows [area_m, 1, 0, 0]. Lanes 0-15 hold K=0,1; lanes 16-31 K=2,3.
    v2f a;
    a.x = lo ? s_area[m0 + (lane & 15)] : 0.0f;
    a.y = lo ? 1.0f : 0.0f;
    // B (4x16): rows K=0 -> ones, K=1 -> area_n, K=2,3 -> 0.
    v2f bm;
    bm.x = lo ? 1.0f : 0.0f;
    bm.y = lo ? s_area[n0 + (lane & 15)] : 0.0f;

    v8f c = {};
    // D[m][n] = area_m + area_n
    v8f d = __builtin_amdgcn_wmma_f32_16x16x4_f32(
        false, a, false, bm, (short)0, c, false, false);

    const int N = n0 + (lane & 15);
    const float nx0 = s_box[N * 4 + 0], ny0 = s_box[N * 4 + 1];
    const float nx1 = s_box[N * 4 + 2], ny1 = s_box[N * 4 + 3];
#pragma unroll
    for (int v = 0; v < 8; ++v) {
      const int M = m0 + v + ((lane >> 4) << 3);
      const float mx0 = s_box[M * 4 + 0], my0 = s_box[M * 4 + 1];
      const float mx1 = s_box[M * 4 + 2], my1 = s_box[M * 4 + 3];
      const float iw = fminf(mx1, nx1) - fmaxf(mx0, nx0);
      const float ih = fminf(my1, ny1) - fmaxf(my0, ny0);
      const float inter = fmaxf(iw, 0.0f) * fmaxf(ih, 0.0f);
      const float uni   = fmaxf(d[v] - inter, 1e-9f);   // union = A_m + A_n - inter
      if ((inter > IOU_THR * uni) && (N > M) && (N < NTOP) && (M < NTOP))
        atomicOr(&s_mask[M * NW + (N >> 5)], 1u << (N & 31));
    }
  }
  __syncthreads();

  // ---- sequential suppression over bitmask rows (threads 0..9 of wave 0) ----
  if (t < NW) {
    unsigned keepw = (t < NW - 1) ? 0xFFFFFFFFu
                                  : ((1u << (NTOP - 32 * (NW - 1))) - 1u);
    for (int i = 0; i < NTOP; ++i) {
      const unsigned kw = __shfl(keepw, i >> 5, 32);
      if ((kw >> (i & 31)) & 1u) keepw &= ~s_mask[i * NW + t];
    }
    s_keep[t] = keepw;
  }
  __syncthreads();

  // ---- masked outputs: scores | labels | boxes | keep ----
  if (t < NTOP) {
    const unsigned kb = (s_keep[t >> 5] >> (t & 31)) & 1u;
    const float keepf = kb ? 1.0f : 0.0f;
    const size_t base = (size_t)b * NTOP + t;
    float* o_scores = out;
    float* o_labels = out + (size_t)NB * NTOP;
    float* o_boxes  = out + (size_t)NB * NTOP * 2;
    float* o_keep   = out + (size_t)NB * NTOP * 6;

    o_scores[base] = kb ? ws_scores[base] : 0.0f;
    o_labels[base] = keepf;                 // label==1 when kept
    float* ob = o_boxes + base * 4;
    ob[0] = kb ? s_box[t * 4 + 0] : 0.0f;
    ob[1] = kb ? s_box[t * 4 + 1] : 0.0f;
    ob[2] = kb ? s_box[t * 4 + 2] : 0.0f;
    ob[3] = kb ? s_box[t * 4 + 3] : 0.0f;
    o_keep[base] = keepf;
  }
}

// =====================================================================
extern "C" void kernel_launch(void* const* d_in, const int* in_sizes, int n_in,
                              void* d_out, int out_size, void* d_ws, size_t ws_size,
                              hipStream_t stream) {
  const float* logits = (const float*)d_in[0];   // [256,16384,2] f32
  const float* pboxes = (const float*)d_in[1];   // [256,16384,4] f32
  const float* tsizes = (const float*)d_in[2];   // [256,2] f32
  float* out = (float*)d_out;

  float* ws_boxes  = (float*)d_ws;                           // B*NTOP*4 floats
  float* ws_scores = ws_boxes + (size_t)NB * NTOP * 4;       // B*NTOP floats

  topk_kernel<<<NB, TPB1, 0, stream>>>(logits, pboxes, tsizes, ws_scores, ws_boxes);
  nms_kernel<<<NB, TPB2, 0, stream>>>(ws_boxes, ws_scores, out);
}